// RelationGraphConv_86706799772303
// MI455X (gfx1250) — compile-verified
//
#include <hip/hip_runtime.h>
#include <math.h>

#define N_SRC_C   50000
#define N_DST_C   50000
#define NEDGE_C   800000
#define D_IN_C    256
#define D_REL_C   64
#define KH_C      8
#define D_OUT_C   32
#define KD_C      (KH_C * D_OUT_C)   /* 256 */
#define NEG_SLOPE_C 0.2f

typedef __attribute__((ext_vector_type(2))) float v2f;
typedef __attribute__((ext_vector_type(8))) float v8f;

// ---------- atomic helpers (lower to single global_atomic_* ops) ----------
__device__ inline void atomic_add_f32(float* p, float v) {
  __hip_atomic_fetch_add(p, v, __ATOMIC_RELAXED, __HIP_MEMORY_SCOPE_AGENT);
}
// float max via monotone integer encodings (emax initialized to -inf)
__device__ inline void atomic_max_f32(float* p, float v) {
  if (v >= 0.0f)
    __hip_atomic_fetch_max((int*)p, __float_as_int(v),
                           __ATOMIC_RELAXED, __HIP_MEMORY_SCOPE_AGENT);
  else
    __hip_atomic_fetch_min((unsigned int*)p, (unsigned int)__float_as_int(v),
                           __ATOMIC_RELAXED, __HIP_MEMORY_SCOPE_AGENT);
}

// ---------- fp32 WMMA GEMM: C(N x 256) = A(N x 256) * W(256 x 256) + b ----------
// One wave computes a 16x32 tile with two v_wmma_f32_16x16x4_f32 accumulators.
__global__ __launch_bounds__(128)
void proj_gemm_wmma(const float* __restrict__ A, const float* __restrict__ W,
                    const float* __restrict__ bias, float* __restrict__ C, int N) {
  const int wave   = threadIdx.x >> 5;
  const int lane   = threadIdx.x & 31;
  const int lane16 = lane & 15;
  const int half   = lane >> 4;          // 0: K=0,1  1: K=2,3  (A frag layout)
  const int row0   = blockIdx.x * 16;
  const int col0   = (blockIdx.y * 4 + wave) * 32;
  if (row0 >= N) return;

  v8f c0 = {};
  v8f c1 = {};
  const float* arow = A + (size_t)(row0 + lane16) * D_IN_C + half * 2;
  const int cA = col0 + lane16;
  const int cB = col0 + 16 + lane16;

#pragma unroll 4
  for (int kk = 0; kk < D_IN_C; kk += 4) {
    // A 16x4 f32 frag: VGPR0 = K=(0|2), VGPR1 = K=(1|3) per half-wave
    v2f a = *(const v2f*)(arow + kk);
    // B 4x16 f32 frag mirrored: lanes 0-15 rows K+0/K+1, lanes 16-31 rows K+2/K+3
    const float* wp = W + (size_t)(kk + 2 * half) * KD_C;
    v2f b0, b1;
    b0.x = wp[cA];        b0.y = wp[KD_C + cA];
    b1.x = wp[cB];        b1.y = wp[KD_C + cB];
    c0 = __builtin_amdgcn_wmma_f32_16x16x4_f32(false, a, false, b0, (short)0, c0, false, false);
    c1 = __builtin_amdgcn_wmma_f32_16x16x4_f32(false, a, false, b1, (short)0, c1, false, false);
  }

  const float biasA = bias[cA], biasB = bias[cB];
#pragma unroll
  for (int j = 0; j < 8; ++j) {           // C layout: VGPR j -> row j (+8 for upper half)
    int r = row0 + half * 8 + j;
    C[(size_t)r * KD_C + cA] = c0[j] + biasA;
    C[(size_t)r * KD_C + cB] = c1[j] + biasB;
  }
}

// ---------- attn = feat_rel @ W_rel + b_rel  (512 outputs) ----------
__global__ void rel_attn_kernel(const float* __restrict__ fr, const float* __restrict__ Wr,
                                const float* __restrict__ br, float* __restrict__ attn) {
  __shared__ float s_fr[D_REL_C];
  if (threadIdx.x < D_REL_C) s_fr[threadIdx.x] = fr[threadIdx.x];
  __syncthreads();
  int o = blockIdx.x * blockDim.x + threadIdx.x;
  if (o < 2 * KD_C) {
    float s = br[o];
#pragma unroll 8
    for (int i = 0; i < D_REL_C; ++i) s += s_fr[i] * Wr[(size_t)i * (2 * KD_C) + o];
    attn[o] = s;
  }
}

// ---------- el/er: per-(node,head) dot of length 32 ----------
__global__ void node_score_kernel(const float* __restrict__ F, const float* __restrict__ attn,
                                  float* __restrict__ score, int N, int attn_off) {
  int t = blockIdx.x * blockDim.x + threadIdx.x;
  if (t >= N * KH_C) return;
  int n = t >> 3, k = t & 7;
  const float* f = F + (size_t)n * KD_C + k * D_OUT_C;
  const float* a = attn + k * (2 * D_OUT_C) + attn_off;
  float s = 0.0f;
#pragma unroll
  for (int d = 0; d < D_OUT_C; d += 4) {
    float4 fv = *(const float4*)(f + d);
    float4 av = *(const float4*)(a + d);
    s += fv.x * av.x + fv.y * av.y + fv.z * av.z + fv.w * av.w;
  }
  score[t] = s;
}

// ---------- init: emax=-inf, esum=0, out=0 ----------
__global__ void init_kernel(float* __restrict__ emax, float* __restrict__ esum,
                            float* __restrict__ out, int nk, int nout) {
  int t = blockIdx.x * blockDim.x + threadIdx.x;
  if (t < nk) { emax[t] = -INFINITY; esum[t] = 0.0f; }
  if (t < nout) out[t] = 0.0f;
}

// ---------- edge pass 1: e = leaky(el[src]+er[dst]); segment max ----------
__global__ void edge_score_max_kernel(const int* __restrict__ src, const int* __restrict__ dst,
                                      const float* __restrict__ el, const float* __restrict__ er,
                                      float* __restrict__ ews, float* __restrict__ emax, int ek) {
  int t = blockIdx.x * blockDim.x + threadIdx.x;
  if (t >= ek) return;
  int e = t >> 3, k = t & 7;
  int s = src[e], d = dst[e];
  float v = el[s * KH_C + k] + er[d * KH_C + k];
  v = (v >= 0.0f) ? v : NEG_SLOPE_C * v;
  ews[t] = v;
  atomic_max_f32(&emax[d * KH_C + k], v);
}

// ---------- edge pass 2: ee = exp(e - emax[dst]); segment sum ----------
__global__ void edge_softmax_kernel(const int* __restrict__ dst, float* __restrict__ ews,
                                    const float* __restrict__ emax, float* __restrict__ esum, int ek) {
  int t = blockIdx.x * blockDim.x + threadIdx.x;
  if (t >= ek) return;
  int e = t >> 3, k = t & 7;
  int d = dst[e];
  float x = __expf(ews[t] - emax[d * KH_C + k]);
  ews[t] = x;
  atomic_add_f32(&esum[d * KH_C + k], x);
}

// ---------- edge pass 3: one wave32 per (edge,head), lane = d ----------
__global__ void edge_aggregate_kernel(const int* __restrict__ src, const int* __restrict__ dst,
                                      const float* __restrict__ fs, const float* __restrict__ ews,
                                      const float* __restrict__ esum, float* __restrict__ out, int ek) {
  int gtid = blockIdx.x * blockDim.x + threadIdx.x;
  int wv   = gtid >> 5;
  int lane = gtid & 31;
  if (wv >= ek) return;
  int e = wv >> 3, k = wv & 7;
  int s = src[e], d = dst[e];
  float a = ews[wv] / esum[d * KH_C + k];
  float m = fs[(size_t)s * KD_C + k * D_OUT_C + lane] * a;
  atomic_add_f32(&out[(size_t)d * KD_C + k * D_OUT_C + lane], m);
}

extern "C" void kernel_launch(void* const* d_in, const int* in_sizes, int n_in,
                              void* d_out, int out_size, void* d_ws, size_t ws_size,
                              hipStream_t stream) {
  (void)in_sizes; (void)n_in; (void)out_size; (void)ws_size;
  const float* feat_src = (const float*)d_in[0];
  const float* feat_dst = (const float*)d_in[1];
  const float* feat_rel = (const float*)d_in[2];
  const int*   src_idx  = (const int*)  d_in[3];
  const int*   dst_idx  = (const int*)  d_in[4];
  const float* W_src    = (const float*)d_in[5];
  const float* b_src    = (const float*)d_in[6];
  const float* W_dst    = (const float*)d_in[7];
  const float* b_dst    = (const float*)d_in[8];
  const float* W_rel    = (const float*)d_in[9];
  const float* b_rel    = (const float*)d_in[10];
  float* out = (float*)d_out;

  // workspace carve (floats)
  float* ws = (float*)d_ws;
  size_t off = 0;
  float* fs   = ws + off; off += (size_t)N_SRC_C * KD_C;   // 12.8M
  float* fd   = ws + off; off += (size_t)N_DST_C * KD_C;   // 12.8M
  float* attn = ws + off; off += 1024;                     // 512 used
  float* el   = ws + off; off += (size_t)N_SRC_C * KH_C;   // 400K
  float* er   = ws + off; off += (size_t)N_DST_C * KH_C;
  float* emax = ws + off; off += (size_t)N_DST_C * KH_C;
  float* esum = ws + off; off += (size_t)N_DST_C * KH_C;
  float* ews  = ws + off; off += (size_t)NEDGE_C * KH_C;   // 6.4M

  const int EK = NEDGE_C * KH_C;  // 6,400,000

  // 1-2: fp32 WMMA projections
  proj_gemm_wmma<<<dim3(N_SRC_C / 16, 2), dim3(128), 0, stream>>>(feat_src, W_src, b_src, fs, N_SRC_C);
  proj_gemm_wmma<<<dim3(N_DST_C / 16, 2), dim3(128), 0, stream>>>(feat_dst, W_dst, b_dst, fd, N_DST_C);
  // 3: attention vector
  rel_attn_kernel<<<dim3(2), dim3(256), 0, stream>>>(feat_rel, W_rel, b_rel, attn);
  // 4-5: node scores
  node_score_kernel<<<dim3((N_SRC_C * KH_C + 255) / 256), dim3(256), 0, stream>>>(fs, attn, el, N_SRC_C, 0);
  node_score_kernel<<<dim3((N_DST_C * KH_C + 255) / 256), dim3(256), 0, stream>>>(fd, attn, er, N_DST_C, D_OUT_C);
  // 6: init accumulators + output
  init_kernel<<<dim3((N_DST_C * KD_C + 255) / 256), dim3(256), 0, stream>>>(
      emax, esum, out, N_DST_C * KH_C, N_DST_C * KD_C);
  // 7: segment max
  edge_score_max_kernel<<<dim3((EK + 255) / 256), dim3(256), 0, stream>>>(
      src_idx, dst_idx, el, er, ews, emax, EK);
  // 8: exp + segment sum
  edge_softmax_kernel<<<dim3((EK + 255) / 256), dim3(256), 0, stream>>>(dst_idx, ews, emax, esum, EK);
  // 9: normalize + weighted scatter-aggregate (wave per (edge,head))
  edge_aggregate_kernel<<<dim3((unsigned)(((size_t)EK * 32 + 255) / 256)), dim3(256), 0, stream>>>(
      src_idx, dst_idx, fs, ews, esum, out, EK);
}